// BQNN_hardware_29686813950726
// MI455X (gfx1250) — compile-verified
//
#include <hip/hip_runtime.h>
#include <math.h>

#define TPB 256
#define AS1 __attribute__((address_space(1)))
#define AS3 __attribute__((address_space(3)))

typedef int v4i __attribute__((vector_size(4 * sizeof(int))));

// ---- CDNA5 async global->LDS copy (ASYNCcnt path) -------------------------
// 16 bytes per lane per op, fully coalesced. Builtin signature (from hipcc
// diagnostic): (v4i AS1* src, v4i AS3* dst, imm offset, imm cpol).
__device__ __forceinline__ void async_ld_b128(const float* base, unsigned byteOff,
                                              float* ldsDst) {
#if __has_builtin(__builtin_amdgcn_global_load_async_to_lds_b128)
    __builtin_amdgcn_global_load_async_to_lds_b128(
        (AS1 v4i*)((char*)base + byteOff),
        (AS3 v4i*)ldsDst, /*offset=*/0, /*cpol=*/0);
#else
    unsigned loff = (unsigned)(unsigned long long)(AS3 float*)ldsDst;
    asm volatile("global_load_async_to_lds_b128 %0, %1, %2"
                 :: "v"(loff), "v"(byteOff), "s"(base) : "memory");
#endif
}

__device__ __forceinline__ void wait_async0() {
#if __has_builtin(__builtin_amdgcn_s_wait_asynccnt)
    __builtin_amdgcn_s_wait_asynccnt(0);
#else
    asm volatile("s_wait_asynccnt 0x0" ::: "memory");
#endif
}

// ---------------------------------------------------------------------------
__global__ __launch_bounds__(TPB) void bqnn_kernel(
    const float* __restrict__ x,      // (B,12)
    const float* __restrict__ pphi,   // (4,)
    const float* __restrict__ pth,    // (8,)
    const float* __restrict__ kk,     // (12,)
    const float* __restrict__ bb,     // (12,)
    float* __restrict__ out,          // (B,15)
    int B)
{
    __shared__ float s_x[TPB * 12];   // 12 KB input stage
    __shared__ float s_o[TPB * 15];   // 15 KB output stage

    const int      tid  = threadIdx.x;
    const unsigned blk  = blockIdx.x;
    const unsigned nV4  = (unsigned)B * 3u;       // total 16B chunks in x
    const unsigned bas4 = blk * (TPB * 3u);       // this block's first chunk

    // DMA tile into LDS: 3 x b128 per lane, coalesced.
#pragma unroll
    for (int q = 0; q < 3; ++q) {
        unsigned e = (unsigned)tid + (unsigned)q * TPB;
        if (bas4 + e < nV4)
            async_ld_b128(x, (bas4 + e) * 16u, &s_x[e * 4u]);
    }
    wait_async0();
    __syncthreads();

    const long long item = (long long)blk * TPB + tid;
    if (item < B) {
        // xs = x*k + b (k,b,params are uniform -> scalar loads)
        float xs[12];
#pragma unroll
        for (int c = 0; c < 12; ++c)
            xs[c] = s_x[tid * 12 + c] * kk[c] + bb[c];

        float pp[4], pt8[8];
#pragma unroll
        for (int c = 0; c < 4; ++c) pp[c] = pphi[c];
#pragma unroll
        for (int c = 0; c < 8; ++c) pt8[c] = pth[c];

        const float th[14] = { pt8[0], pt8[1], pt8[2], pt8[3],
                               xs[3],  xs[4],  xs[5],
                               pt8[4], pt8[5],
                               xs[9],  xs[10], xs[11],
                               pt8[6], pt8[7] };
        const float ph[14] = { 0.f, 0.f, 0.f, 0.f,
                               xs[0], xs[1], xs[2],
                               pp[0], pp[1],
                               xs[6], xs[7], xs[8],
                               pp[2], pp[3] };

        constexpr int MI[14] = {0,4,1,3, 0,2,4,1,3, 0,2,4,1,3};
        constexpr int MJ[14] = {1,5,2,4, 1,3,5,2,4, 1,3,5,2,4};

        // Only columns 0 and 3 of U are needed: propagate a = U e0, b = U e3.
        float ar[6] = {1.f,0,0,0,0,0}, ai[6] = {0,0,0,0,0,0};
        float br[6] = {0,0,0,1.f,0,0}, bi[6] = {0,0,0,0,0,0};

#pragma unroll
        for (int k = 0; k < 14; ++k) {
            const int i = MI[k], j = MJ[k];
            float st, ct;  __sincosf(th[k], &st, &ct);
            float er, ei;
            if (k < 4) { er = 1.f; ei = 0.f; }        // phi == 0 exactly
            else       { __sincosf(ph[k], &ei, &er); }
            const float pr = er * ct, pi = ei * ct;   // ep*cos(t)
            const float qr = er * st, qi = ei * st;   // ep*sin(t)

            {   // rotate column-0 vector
                const float vir = ar[i], vii = ai[i], vjr = ar[j], vji = ai[j];
                ar[i] = pr * vir - pi * vii - st * vjr;
                ai[i] = pr * vii + pi * vir - st * vji;
                ar[j] = qr * vir - qi * vii + ct * vjr;
                ai[j] = qr * vii + qi * vir + ct * vji;
            }
            {   // rotate column-3 vector
                const float vir = br[i], vii = bi[i], vjr = br[j], vji = bi[j];
                br[i] = pr * vir - pi * vii - st * vjr;
                bi[i] = pr * vii + pi * vir - st * vji;
                br[j] = qr * vir - qi * vii + ct * vjr;
                bi[j] = qr * vii + qi * vir + ct * vji;
            }
        }

        constexpr int OA[15] = {0,0,0,0,0,1,1,1,1,2,2,2,3,3,4};
        constexpr int OB[15] = {1,2,3,4,5,2,3,4,5,3,4,5,4,5,5};

        float m[15];
        float sum = 0.f;
#pragma unroll
        for (int p = 0; p < 15; ++p) {
            const int i = OA[p], j = OB[p];
            const float re = ar[i]*br[j] - ai[i]*bi[j] + br[i]*ar[j] - bi[i]*ai[j];
            const float im = ar[i]*bi[j] + ai[i]*br[j] + br[i]*ai[j] + bi[i]*ar[j];
            const float m2 = re * re + im * im;
            sum += m2;
            m[p] = __builtin_sqrtf(m2);
        }
        const float inv = 1.0f / fmaxf(__builtin_sqrtf(sum), 1e-12f);
#pragma unroll
        for (int p = 0; p < 15; ++p) s_o[tid * 15 + p] = m[p] * inv;
    }
    __syncthreads();

    // Coalesced flush of the (TPB x 15) tile.
    const long long outBase  = (long long)blk * (TPB * 15);
    const long long outTotal = (long long)B * 15;
    for (int i2 = tid; i2 < TPB * 15; i2 += TPB) {
        const long long o = outBase + i2;
        if (o < outTotal) out[o] = s_o[i2];
    }
}

extern "C" void kernel_launch(void* const* d_in, const int* in_sizes, int n_in,
                              void* d_out, int out_size, void* d_ws, size_t ws_size,
                              hipStream_t stream) {
    const float* x    = (const float*)d_in[0];
    const float* pphi = (const float*)d_in[1];
    const float* pth  = (const float*)d_in[2];
    const float* kk   = (const float*)d_in[3];
    const float* bb   = (const float*)d_in[4];
    float* out = (float*)d_out;

    const int B = in_sizes[0] / 12;
    const int blocks = (B + TPB - 1) / TPB;
    hipLaunchKernelGGL(bqnn_kernel, dim3(blocks), dim3(TPB), 0, stream,
                       x, pphi, pth, kk, bb, out, B);
}